// EgoMotionHead_35854386987669
// MI455X (gfx1250) — compile-verified
//
#include <hip/hip_runtime.h>
#include <math.h>

#define NB 16
#define NJ 2048
#define NK 2048
#define JSPLIT 8

typedef float v2f __attribute__((ext_vector_type(2)));
typedef float v4f __attribute__((ext_vector_type(4)));
typedef float v8f __attribute__((ext_vector_type(8)));
typedef unsigned int v4u __attribute__((ext_vector_type(4)));
typedef int v4i __attribute__((ext_vector_type(4)));
typedef int v8i __attribute__((ext_vector_type(8)));

__device__ __forceinline__ float wave_sum(float v) {
#pragma unroll
  for (int off = 16; off > 0; off >>= 1) v += __shfl_xor(v, off, 32);
  return v;
}

__device__ __forceinline__ void sinkhorn_params(const float* pa, const float* pb,
                                                float& sp, float& invd) {
  float a = pa[0], b = pb[0];
  sp = __logf(1.0f + __expf(a));       // softplus(alpha)
  invd = 1.0f / (__expf(b) + 0.02f);   // 1 / (exp(beta) + 0.02)
}

// ---------------- c := 0 ----------------
__global__ void init_c_kernel(float* c) {
  int i = blockIdx.x * blockDim.x + threadIdx.x;
  if (i < NB * NK) c[i] = 0.0f;
}

// ------- row step: r_j = log(1 + sum_k exp(aff - c_k)); one wave per row -------
__global__ __launch_bounds__(256) void row_pass_kernel(
    const float* __restrict__ score, const float* __restrict__ c,
    float* __restrict__ r, const float* __restrict__ pa, const float* __restrict__ pb) {
  int b = blockIdx.x;
  int wave = threadIdx.x >> 5, lane = threadIdx.x & 31;
  int j = blockIdx.y * 8 + wave;
  float sp, invd;
  sinkhorn_params(pa, pb, sp, invd);
  const v4f* srow = (const v4f*)(score + ((size_t)b * NJ + j) * NK);
  const v4f* crow = (const v4f*)(c + (size_t)b * NK);
  float sum = 0.0f;
#pragma unroll 4
  for (int i = lane; i < NK / 4; i += 32) {
    v4f s = srow[i];
    v4f cc = crow[i];
    sum += __expf((sp - s.x) * invd - cc.x);
    sum += __expf((sp - s.y) * invd - cc.y);
    sum += __expf((sp - s.z) * invd - cc.z);
    sum += __expf((sp - s.w) * invd - cc.w);
  }
  sum = wave_sum(sum);
  if (lane == 0) r[b * NJ + j] = __logf(1.0f + sum);
}

// ------- col step (partials): thread-per-column over a J stripe -------
__global__ __launch_bounds__(256) void col_partial_kernel(
    const float* __restrict__ score, const float* __restrict__ r,
    float* __restrict__ partial, const float* __restrict__ pa,
    const float* __restrict__ pb) {
  int b = blockIdx.x;
  int k = blockIdx.y * 256 + threadIdx.x;
  int j0 = blockIdx.z * (NJ / JSPLIT);
  float sp, invd;
  sinkhorn_params(pa, pb, sp, invd);
  const float* scol = score + (size_t)b * NJ * NK + k;
  const float* rb = r + b * NJ;
  float sum = 0.0f;
  for (int j = j0; j < j0 + NJ / JSPLIT; j += 8) {
    __builtin_prefetch(scol + (size_t)(j + 32) * NK, 0, 1);
#pragma unroll
    for (int u = 0; u < 8; ++u)
      sum += __expf((sp - scol[(size_t)(j + u) * NK]) * invd - rb[j + u]);
  }
  partial[(size_t)(b * JSPLIT + blockIdx.z) * NK + k] = sum;
}

// ------- col step (finalize): c_k = log(1 + sum of stripe partials) -------
__global__ void col_finalize_kernel(const float* __restrict__ partial,
                                    float* __restrict__ c) {
  int idx = blockIdx.x * blockDim.x + threadIdx.x;
  if (idx >= NB * NK) return;
  int b = idx >> 11, k = idx & (NK - 1);
  float s = 0.0f;
#pragma unroll
  for (int z = 0; z < JSPLIT; ++z) s += partial[(size_t)(b * JSPLIT + z) * NK + k];
  c[idx] = __logf(1.0f + s);
}

// ------- final fused pass: perm out + WMMA perm @ [x y z 1] -------
// One block = 16 rows x full K. 8 waves each own a K/8 slice.
// A (16x4 f32) = perm tile, B (4x16) = [x;y;z;1] columns; D cols 0..3 used
// (lanes n>=4 read duplicate B columns -> D cols 4..15 are ignored duplicates).
__global__ __launch_bounds__(256) void final_pass_kernel(
    const float* __restrict__ score, const float* __restrict__ mask,
    const float* __restrict__ xyz_t, const float* __restrict__ r,
    const float* __restrict__ c, const float* __restrict__ pa,
    const float* __restrict__ pb, float* __restrict__ out,
    float* __restrict__ sums4) {
  __shared__ float ldsXYZ[NK * 3];   // raw xyz_t[b], staged by TDM
  __shared__ float bmat[NK * 4];     // packed {x,y,z,1} per k
  __shared__ float red[8 * 16 * 4];
  int b = blockIdx.x;
  int jb = blockIdx.y * 16;
  int tid = threadIdx.x;
  int wave = __builtin_amdgcn_readfirstlane(tid >> 5);  // force SGPR (uniform/wave)
  int lane = tid & 31;
  int hf = lane >> 4, n = lane & 15;

  // ---- stage xyz_t[b] (6144 contiguous f32) into LDS via Tensor Data Mover ----
#if __has_builtin(__builtin_amdgcn_tensor_load_to_lds)
  if (wave == 0) {
    // Hand-built Tensor DMA Descriptor (CDNA5 ISA):
    // 1-D tile: data_size=4B, tensor_dim0 = tile_dim0 = 6144, dim1 = tile1 = 1.
    unsigned long long ga =
        (unsigned long long)(uintptr_t)(xyz_t + (size_t)b * NK * 3);
    unsigned int la = (unsigned int)(uintptr_t)(&ldsXYZ[0]);  // LDS byte offset
    v4u g0 = {1u,                                   // count=1, user mode
              la,                                   // lds_addr (bits 63:32)
              (unsigned int)ga,                     // global_addr[31:0]
              (unsigned int)((ga >> 32) & 0x01FFFFFFu) | 0x80000000u};  // type=2
    v8i g1 = {(int)0x00020000,                      // data_size = 2 (4 bytes)
              (int)((NK * 3) << 16),                // tensor_dim0 (bits 79:48) low 16
              (int)0x00010000,                      // tensor_dim1 = 1
              (int)((NK * 3) << 16),                // tile_dim0 = 6144
              (int)0x00000001,                      // tile_dim1 = 1, tile_dim2 = 0
              (int)(NK * 3),                        // tensor_dim0_stride = 6144
              0, 0};
    v4i z4 = {0, 0, 0, 0};
    v8i z8 = {0, 0, 0, 0, 0, 0, 0, 0};
    __builtin_amdgcn_tensor_load_to_lds(g0, g1, z4, z4, z8, 0);
    __builtin_amdgcn_s_wait_tensorcnt(0);
  }
#else
  for (int i = tid; i < NK * 3; i += 256) ldsXYZ[i] = xyz_t[(size_t)b * NK * 3 + i];
#endif
  __syncthreads();
  // repack to {x,y,z,1} stride-4 layout for branch-free B-operand loads
  for (int k = tid; k < NK; k += 256) {
    bmat[k * 4 + 0] = ldsXYZ[k * 3 + 0];
    bmat[k * 4 + 1] = ldsXYZ[k * 3 + 1];
    bmat[k * 4 + 2] = ldsXYZ[k * 3 + 2];
    bmat[k * 4 + 3] = 1.0f;
  }
  __syncthreads();

  float sp, invd;
  sinkhorn_params(pa, pb, sp, invd);

  int row = jb + n;  // A-matrix row for this lane (both halves)
  float rrow = r[b * NJ + row];
  const float* srow = score + ((size_t)b * NJ + row) * NK;
  const float* mrow = mask + ((size_t)b * NJ + row) * NK;
  float* prow = out + 192 + ((size_t)b * NJ + row) * NK;  // perm region
  const float* cb = c + (size_t)b * NK;
  int bcol = n & 3;  // B column this lane supplies (n>=4 duplicates 0..3)

  v8f acc = {0.f, 0.f, 0.f, 0.f, 0.f, 0.f, 0.f, 0.f};
  int kbeg = wave * (NK / 8);
  for (int k0 = kbeg; k0 < kbeg + NK / 8; k0 += 8) {
    int ka = k0 + 4 * hf;  // this lane's float4 start (half0: k0.., half1: k0+4..)
    v4f s = *(const v4f*)(srow + ka);
    v4f m = *(const v4f*)(mrow + ka);
    v4f cc = *(const v4f*)(cb + ka);
    v4f p;
    p.x = __expf((sp - s.x) * invd - rrow - cc.x) * m.x;
    p.y = __expf((sp - s.y) * invd - rrow - cc.y) * m.y;
    p.z = __expf((sp - s.z) * invd - rrow - cc.z) * m.z;
    p.w = __expf((sp - s.w) * invd - rrow - cc.w) * m.w;
    __builtin_nontemporal_store(p, (v4f*)(prow + ka));

    // branch-free B operand from packed LDS
    int bi = ka * 4 + bcol;
    float b0 = bmat[bi + 0];
    float b1 = bmat[bi + 4];
    float b2 = bmat[bi + 8];
    float b3 = bmat[bi + 12];

    // wmma #1: K-set {k0,k0+1,k0+4,k0+5}
    v2f A1 = {p.x, p.y}, B1 = {b0, b1};
    acc = __builtin_amdgcn_wmma_f32_16x16x4_f32(false, A1, false, B1, (short)0, acc,
                                                false, false);
    // wmma #2: K-set {k0+2,k0+3,k0+6,k0+7}
    v2f A2 = {p.z, p.w}, B2 = {b2, b3};
    acc = __builtin_amdgcn_wmma_f32_16x16x4_f32(false, A2, false, B2, (short)0, acc,
                                                false, false);
  }

  // D layout: vgpr v, lane<16 -> (M=v, N=lane); lane>=16 -> (M=v+8, N=lane-16)
  if (n < 4 && (lane < 4 || lane >= 16)) {
#pragma unroll
    for (int v = 0; v < 8; ++v)
      red[(wave * 16 + (v + (hf ? 8 : 0))) * 4 + n] = acc[v];
  }
  __syncthreads();
  if (tid < 64) {
    int m = tid >> 2, nn = tid & 3;
    float t = 0.0f;
#pragma unroll
    for (int w = 0; w < 8; ++w) t += red[(w * 16 + m) * 4 + nn];
    sums4[((size_t)b * NJ + jb + m) * 4 + nn] = t;  // (sx, sy, sz, rowsum)
  }
}

// ------- weighted Procrustes per batch -------
__device__ __forceinline__ float block_sum(float v, float* sdata, int tid) {
  __syncthreads();
  sdata[tid] = v;
  __syncthreads();
  for (int s = 128; s > 0; s >>= 1) {
    if (tid < s) sdata[tid] += sdata[tid + s];
    __syncthreads();
  }
  float r = sdata[0];
  __syncthreads();
  return r;
}

__global__ __launch_bounds__(256) void rigid_kernel(const float* __restrict__ xyz_s,
                                                    const float* __restrict__ sums4,
                                                    float* __restrict__ out) {
  __shared__ float sdata[256];
  int b = blockIdx.x, tid = threadIdx.x;
  float W = 0, Ss0 = 0, Ss1 = 0, Ss2 = 0, St0 = 0, St1 = 0, St2 = 0;
  for (int j = tid; j < NJ; j += 256) {
    const float* q = sums4 + ((size_t)b * NJ + j) * 4;
    float w = q[3];
    float inv = 1.0f / (w + 1e-6f);
    float t0 = q[0] * inv, t1 = q[1] * inv, t2 = q[2] * inv;
    const float* s = xyz_s + ((size_t)b * NJ + j) * 3;
    W += w;
    Ss0 += w * s[0]; Ss1 += w * s[1]; Ss2 += w * s[2];
    St0 += w * t0;   St1 += w * t1;   St2 += w * t2;
  }
  W = block_sum(W, sdata, tid);
  Ss0 = block_sum(Ss0, sdata, tid); Ss1 = block_sum(Ss1, sdata, tid);
  Ss2 = block_sum(Ss2, sdata, tid);
  St0 = block_sum(St0, sdata, tid); St1 = block_sum(St1, sdata, tid);
  St2 = block_sum(St2, sdata, tid);
  float invW = 1.0f / (W + 1e-6f);
  float cs0 = Ss0 * invW, cs1 = Ss1 * invW, cs2 = Ss2 * invW;
  float ct0 = St0 * invW, ct1 = St1 * invW, ct2 = St2 * invW;

  float E00 = 0, E01 = 0, E02 = 0, E10 = 0, E11 = 0, E12 = 0, E20 = 0, E21 = 0, E22 = 0;
  for (int j = tid; j < NJ; j += 256) {
    const float* q = sums4 + ((size_t)b * NJ + j) * 4;
    float w = q[3];
    float inv = 1.0f / (w + 1e-6f);
    float dt0 = q[0] * inv - ct0, dt1 = q[1] * inv - ct1, dt2 = q[2] * inv - ct2;
    const float* s = xyz_s + ((size_t)b * NJ + j) * 3;
    float ds0 = s[0] - cs0, ds1 = s[1] - cs1, ds2 = s[2] - cs2;
    E00 += w * ds0 * dt0; E01 += w * ds0 * dt1; E02 += w * ds0 * dt2;
    E10 += w * ds1 * dt0; E11 += w * ds1 * dt1; E12 += w * ds1 * dt2;
    E20 += w * ds2 * dt0; E21 += w * ds2 * dt1; E22 += w * ds2 * dt2;
  }
  E00 = block_sum(E00, sdata, tid); E01 = block_sum(E01, sdata, tid);
  E02 = block_sum(E02, sdata, tid); E10 = block_sum(E10, sdata, tid);
  E11 = block_sum(E11, sdata, tid); E12 = block_sum(E12, sdata, tid);
  E20 = block_sum(E20, sdata, tid); E21 = block_sum(E21, sdata, tid);
  E22 = block_sum(E22, sdata, tid);

  if (tid == 0) {
    float C[3][3] = {{E00 * invW, E01 * invW, E02 * invW},
                     {E10 * invW, E11 * invW, E12 * invW},
                     {E20 * invW, E21 * invW, E22 * invW}};
    float M[3][3];
    for (int i = 0; i < 3; ++i)
      for (int j = 0; j < 3; ++j)
        M[i][j] = C[0][i] * C[0][j] + C[1][i] * C[1][j] + C[2][i] * C[2][j];
    float V[3][3] = {{1, 0, 0}, {0, 1, 0}, {0, 0, 1}};
    const int pr[3][2] = {{0, 1}, {0, 2}, {1, 2}};
    for (int sweep = 0; sweep < 12; ++sweep) {
      for (int t = 0; t < 3; ++t) {
        int p = pr[t][0], q = pr[t][1];
        float apq = M[p][q];
        if (fabsf(apq) < 1e-20f) continue;
        float tau = (M[q][q] - M[p][p]) / (2.0f * apq);
        float tt = (tau >= 0.0f ? 1.0f : -1.0f) / (fabsf(tau) + sqrtf(1.0f + tau * tau));
        float cth = 1.0f / sqrtf(1.0f + tt * tt);
        float sth = tt * cth;
        for (int kk = 0; kk < 3; ++kk) {
          float a = M[kk][p], bb = M[kk][q];
          M[kk][p] = cth * a - sth * bb;
          M[kk][q] = sth * a + cth * bb;
        }
        for (int kk = 0; kk < 3; ++kk) {
          float a = M[p][kk], bb = M[q][kk];
          M[p][kk] = cth * a - sth * bb;
          M[q][kk] = sth * a + cth * bb;
        }
        for (int kk = 0; kk < 3; ++kk) {
          float a = V[kk][p], bb = V[kk][q];
          V[kk][p] = cth * a - sth * bb;
          V[kk][q] = sth * a + cth * bb;
        }
      }
    }
    float e[3] = {M[0][0], M[1][1], M[2][2]};
    for (int a = 0; a < 2; ++a)
      for (int bb = a + 1; bb < 3; ++bb)
        if (e[bb] > e[a]) {
          float te = e[a]; e[a] = e[bb]; e[bb] = te;
          for (int kk = 0; kk < 3; ++kk) {
            float tv = V[kk][a]; V[kk][a] = V[kk][bb]; V[kk][bb] = tv;
          }
        }
    float U[3][3];
    for (int i = 0; i < 3; ++i) {
      float si = sqrtf(fmaxf(e[i], 0.0f));
      float inv_s = 1.0f / (si + 1e-12f);
      for (int jj = 0; jj < 3; ++jj)
        U[jj][i] = (C[jj][0] * V[0][i] + C[jj][1] * V[1][i] + C[jj][2] * V[2][i]) * inv_s;
    }
    float R[3][3];
    for (int i = 0; i < 3; ++i)
      for (int j = 0; j < 3; ++j)
        R[i][j] = V[i][0] * U[j][0] + V[i][1] * U[j][1] + V[i][2] * U[j][2];
    float det = R[0][0] * (R[1][1] * R[2][2] - R[1][2] * R[2][1]) -
                R[0][1] * (R[1][0] * R[2][2] - R[1][2] * R[2][0]) +
                R[0][2] * (R[1][0] * R[2][1] - R[1][1] * R[2][0]);
    if (!(det > 0.0f)) {
      for (int i = 0; i < 3; ++i)
        for (int j = 0; j < 3; ++j) R[i][j] -= 2.0f * V[i][2] * U[j][2];
    }
    float tr0 = -(R[0][0] * cs0 + R[0][1] * cs1 + R[0][2] * cs2) + ct0;
    float tr1 = -(R[1][0] * cs0 + R[1][1] * cs1 + R[1][2] * cs2) + ct1;
    float tr2 = -(R[2][0] * cs0 + R[2][1] * cs1 + R[2][2] * cs2) + ct2;
    for (int i = 0; i < 3; ++i)
      for (int j = 0; j < 3; ++j) out[b * 9 + i * 3 + j] = R[i][j];
    out[144 + b * 3 + 0] = tr0;
    out[144 + b * 3 + 1] = tr1;
    out[144 + b * 3 + 2] = tr2;
  }
}

extern "C" void kernel_launch(void* const* d_in, const int* in_sizes, int n_in,
                              void* d_out, int out_size, void* d_ws, size_t ws_size,
                              hipStream_t stream) {
  const float* score = (const float*)d_in[0];
  const float* mask = (const float*)d_in[1];
  const float* xyz_s = (const float*)d_in[2];
  const float* xyz_t = (const float*)d_in[3];
  const float* alpha = (const float*)d_in[4];
  const float* beta = (const float*)d_in[5];
  float* out = (float*)d_out;
  float* ws = (float*)d_ws;

  float* c = ws;                 // NB*NK        = 32768 f
  float* r = ws + 32768;         // NB*NJ        = 32768 f
  float* partial = ws + 65536;   // NB*JSPLIT*NK = 262144 f
  float* sums4 = ws + 327680;    // NB*NJ*4      = 131072 f

  init_c_kernel<<<(NB * NK + 255) / 256, 256, 0, stream>>>(c);
  for (int it = 0; it < 5; ++it) {
    row_pass_kernel<<<dim3(NB, NJ / 8), 256, 0, stream>>>(score, c, r, alpha, beta);
    col_partial_kernel<<<dim3(NB, NK / 256, JSPLIT), 256, 0, stream>>>(score, r, partial,
                                                                       alpha, beta);
    col_finalize_kernel<<<(NB * NK + 255) / 256, 256, 0, stream>>>(partial, c);
  }
  final_pass_kernel<<<dim3(NB, NJ / 16), 256, 0, stream>>>(score, mask, xyz_t, r, c,
                                                           alpha, beta, out, sums4);
  rigid_kernel<<<NB, 256, 0, stream>>>(xyz_s, sums4, out);
}